// PoolingLayer_81398220193831
// MI455X (gfx1250) — compile-verified
//
#include <hip/hip_runtime.h>
#include <hip/hip_bf16.h>

// Segment-mean pooling (sorted segment ids), MI455X / gfx1250.
//
//   K1: 1025 binary searches on sorted batch -> segment row ranges (d_ws).
//   K2: one 256-thread block (8 wave32) per segment:
//       - double-buffered ASYNC global->LDS staging (CDNA5
//         global_load_async_to_lds_b128 + s_wait_asynccnt) of 32-row tiles,
//         ragged tail staged via masked b128 loads + ds_store
//       - per wave: 4 rows summed via V_WMMA_F32_16X16X4_F32 with all-ones A
//         (C[m,n] += sum_k B[k,n]; exact f32, layout-of-A independent)
//       - cross-wave LDS reduce, fused divide by count, store mean.
// No atomics, single pass over x, deterministic.

typedef __attribute__((ext_vector_type(2))) float v2f;
typedef __attribute__((ext_vector_type(4))) float v4f;
typedef __attribute__((ext_vector_type(8))) float v8f;

// Async-LDS builtin operand types (signature learned from hipcc diagnostic:
// param0 is '__attribute__((vector_size(16))) int AS(1) *', 4 args total).
typedef __attribute__((vector_size(16))) int v4i;
typedef __attribute__((address_space(1))) v4i v4i_glob;
typedef __attribute__((address_space(3))) v4i v4i_lds;

#define NUM_SEGMENTS 1024
#define FEAT         128
#define TILE_ROWS    32          // rows per tile (4 per wave x 8 waves)
#define LDS_STRIDE   144         // 128 + 16 floats pad -> conflict-free half-wave reads

#if __has_builtin(__builtin_amdgcn_global_load_async_to_lds_b128)
#define HAVE_ASYNC_LDS 1
#else
#define HAVE_ASYNC_LDS 0
#endif

template <int N>
__device__ __forceinline__ void wait_async() {
#if __has_builtin(__builtin_amdgcn_s_wait_asynccnt)
    __builtin_amdgcn_s_wait_asynccnt(N);
#else
    asm volatile("s_wait_asynccnt %0" ::"i"(N) : "memory");
#endif
}

__global__ void seg_starts_kernel(const int* __restrict__ batch, int n,
                                  int* __restrict__ starts) {
    int s = blockIdx.x * blockDim.x + threadIdx.x;
    if (s > NUM_SEGMENTS) return;
    // lower_bound: first i with batch[i] >= s
    int lo = 0, hi = n;
    while (lo < hi) {
        int mid = (lo + hi) >> 1;
        if (batch[mid] < s) lo = mid + 1; else hi = mid;
    }
    starts[s] = lo;
}

// Masked synchronous staging (used for the ragged tail, and as full fallback).
__device__ __forceinline__ void stage_tile_sync(float* tbuf, const float* x,
                                                int rbase, int end, int tid) {
#pragma unroll
    for (int j = 0; j < 4; ++j) {
        int c    = tid + j * 256;        // 0..1023 v4f chunks
        int row  = c >> 5;               // 0..31
        int col4 = c & 31;               // 0..31
        int grow = rbase + row;
        v4f v = (v4f){};
        if (grow < end)
            v = *(const v4f*)(x + (size_t)grow * FEAT + col4 * 4);
        *(v4f*)(tbuf + row * LDS_STRIDE + col4 * 4) = v;
    }
}

#if HAVE_ASYNC_LDS
// Async DMA of a FULL 32-row tile straight into LDS (no VGPR staging).
// 4 async-b128 instructions per wave -> ASYNCcnt +4, completes in order.
__device__ __forceinline__ void issue_tile_async(float* tbuf, const float* x,
                                                 int rbase, int tid) {
#pragma unroll
    for (int j = 0; j < 4; ++j) {
        int c    = tid + j * 256;
        int row  = c >> 5;
        int col4 = c & 31;
        __builtin_amdgcn_global_load_async_to_lds_b128(
            (v4i_glob*)(x + (size_t)(rbase + row) * FEAT + col4 * 4),
            (v4i_lds*)(tbuf + row * LDS_STRIDE + col4 * 4),
            0, 0);
    }
}
#endif

// Fold 4 rows (r0..r0+3) of a staged tile into the 8 WMMA accumulators.
__device__ __forceinline__ void wmma_accum_tile(const float* tbuf, int r0, int kb,
                                                int nn, v2f onesA, v8f acc[8]) {
#pragma unroll
    for (int t = 0; t < 8; ++t) {
        v2f b;
        b.x = tbuf[(r0 + kb    ) * LDS_STRIDE + t * 16 + nn];
        b.y = tbuf[(r0 + 2 + kb) * LDS_STRIDE + t * 16 + nn];
        acc[t] = __builtin_amdgcn_wmma_f32_16x16x4_f32(
            false, onesA, false, b, (short)0, acc[t], false, false);
    }
}

__global__ void __launch_bounds__(256, 4)
seg_mean_kernel(const float* __restrict__ x,
                const int* __restrict__ starts,
                float* __restrict__ out) {
    __shared__ float tile[2][TILE_ROWS * LDS_STRIDE];   // 2 x 18 KB staging
    __shared__ float partial[8 * FEAT];                 // 4 KB cross-wave reduce

    const int seg  = blockIdx.x;
    const int tid  = threadIdx.x;
    const int wave = tid >> 5;
    const int lane = tid & 31;

    // Segment bounds into SGPRs -> scalar loop control, EXEC stays all-ones
    // around the WMMAs (ISA requirement).
    const int begin = __builtin_amdgcn_readfirstlane(starts[seg]);
    const int end   = __builtin_amdgcn_readfirstlane(starts[seg + 1]);
    const int count = end - begin;

    v8f acc[8];
#pragma unroll
    for (int t = 0; t < 8; ++t) acc[t] = (v8f){};

    const v2f onesA = {1.0f, 1.0f};     // all-ones 16x4 A: layout-proof row summer
    const int kb = lane >> 4;           // which B-row this half-wave feeds
    const int nn = lane & 15;           // B column == C column N
    const int r0 = wave * 4;

#if HAVE_ASYNC_LDS
    const int nfull    = count >> 5;                 // full 32-row tiles
    const int tailBase = begin + nfull * TILE_ROWS;

    if (nfull > 0) issue_tile_async(tile[0], x, begin, tid);
    int buf = 0;
    for (int i = 0; i < nfull; ++i) {
        if (i + 1 < nfull) {
            issue_tile_async(tile[buf ^ 1], x, begin + (i + 1) * TILE_ROWS, tid);
            wait_async<4>();            // tile i landed (per-wave, in order)
        } else {
            wait_async<0>();            // last tile landed
        }
        __syncthreads();                // all waves' DMA complete -> tile ready
        wmma_accum_tile(tile[buf], r0, kb, nn, onesA, acc);
        __syncthreads();                // all waves done reading before reuse
        buf ^= 1;
    }
    if (tailBase < end) {               // ragged tail: masked loads, zero-filled
        stage_tile_sync(tile[buf], x, tailBase, end, tid);
        __syncthreads();
        wmma_accum_tile(tile[buf], r0, kb, nn, onesA, acc);
        __syncthreads();
    }
#else
    for (int rbase = begin; rbase < end; rbase += TILE_ROWS) {
        stage_tile_sync(tile[0], x, rbase, end, tid);
        __syncthreads();
        wmma_accum_tile(tile[0], r0, kb, nn, onesA, acc);
        __syncthreads();
    }
#endif

    // Row M=0 of each C holds the column sums (replicated over all M).
    if (lane < 16) {
#pragma unroll
        for (int t = 0; t < 8; ++t)
            partial[wave * FEAT + t * 16 + lane] = acc[t][0];
    }
    __syncthreads();

    if (tid < FEAT) {
        float s = 0.0f;
#pragma unroll
        for (int w = 0; w < 8; ++w) s += partial[w * FEAT + tid];
        float inv = count > 0 ? 1.0f / (float)count : 0.0f;
        out[(size_t)seg * FEAT + tid] = s * inv;
    }
}

extern "C" void kernel_launch(void* const* d_in, const int* in_sizes, int n_in,
                              void* d_out, int out_size, void* d_ws, size_t ws_size,
                              hipStream_t stream) {
    const float* x     = (const float*)d_in[0];
    const int*   batch = (const int*)d_in[1];
    float*       out   = (float*)d_out;
    const int    n     = in_sizes[1];

    int* starts = (int*)d_ws;   // 1025 ints

    seg_starts_kernel<<<(NUM_SEGMENTS + 1 + 255) / 256, 256, 0, stream>>>(batch, n, starts);
    seg_mean_kernel<<<NUM_SEGMENTS, 256, 0, stream>>>(x, starts, out);
}